// FirstHitProjector_88304527605960
// MI455X (gfx1250) — compile-verified
//
#include <hip/hip_runtime.h>

// FirstHitProjector: B=4, C=3, K=128, H=W=256, fp32 logits.
// Per pixel: first depth k where argmax_c(logits[:,k]) != 0; emit that class
// (else 0) and softmax over C at that slice (else at slice K-1).
//
// Memory-bound streaming scan. gfx1250 specifics used:
//  - GVS addressing: wave-uniform SGPR base + 32-bit per-lane voffset
//  - non-temporal loads (th:TH_LOAD_NT) for the read-once 402 MB input
//  - early exit (expected ~3-4 of 128 slices read per wave)

constexpr int       Cc   = 3;
constexpr int       Kk   = 128;
constexpr int       Bb   = 4;
constexpr long long HW   = 256LL * 256LL;        // 65536
constexpr long long KHW  = (long long)Kk * HW;   // 8388608 elems (32 MB)
constexpr long long CKHW = (long long)Cc * KHW;  // 25165824 elems

constexpr unsigned  KSTEP_BYTES = (unsigned)(HW * 4);   // 256 KB per depth slice
constexpr unsigned  CSTR_BYTES  = (unsigned)(KHW * 4);  // 32 MB per class plane

__global__ __launch_bounds__(256)
void first_hit_projector_kernel(const float* __restrict__ V,
                                int* __restrict__ out_cls,
                                float* __restrict__ out_probs)
{
    const int b = blockIdx.y;                               // uniform per block
    const int r = blockIdx.x * blockDim.x + threadIdx.x;    // h*W + w, exact grid

    // Wave-uniform 64-bit base (SGPR) + per-lane 32-bit byte offsets (VGPR)
    // -> GVS global_load form, no per-lane 64-bit address math in the loop.
    const char* __restrict__ base = (const char*)(V + (size_t)b * (size_t)CKHW);

    unsigned o0 = (unsigned)r * 4u;        // class 0, slice k
    unsigned o1 = o0 + CSTR_BYTES;         // class 1
    unsigned o2 = o0 + 2u * CSTR_BYTES;    // class 2  (max ~100 MB, fits u32)

    float s0 = 0.f, s1 = 0.f, s2 = 0.f;    // logits of "current" slice
    int   vcls  = 0;
    bool  found = false;

    #pragma unroll 1
    for (int k = 0; k < Kk; ++k) {
        const float l0 = __builtin_nontemporal_load((const float*)(base + o0));
        const float l1 = __builtin_nontemporal_load((const float*)(base + o1));
        const float l2 = __builtin_nontemporal_load((const float*)(base + o2));

        s0 = l0; s1 = l1; s2 = l2;

        // jnp.argmax tie-break: lowest index wins.
        const int cls = (l0 >= l1 && l0 >= l2) ? 0 : ((l1 >= l2) ? 1 : 2);
        if (cls != 0) { vcls = cls; found = true; break; }

        o0 += KSTEP_BYTES; o1 += KSTEP_BYTES; o2 += KSTEP_BYTES;
    }

    // Single 3-way softmax at the resolved slice (first hit, or K-1 fallback).
    const float m   = fmaxf(s0, fmaxf(s1, s2));
    const float e0  = expf(s0 - m);
    const float e1  = expf(s1 - m);
    const float e2  = expf(s2 - m);
    const float inv = 1.0f / (e0 + e1 + e2);

    out_cls[b * (int)HW + r] = found ? vcls : 0;

    // visible_probs layout (B, C, H, W)
    float* __restrict__ op = out_probs + (size_t)b * (size_t)(Cc * HW) + (size_t)r;
    op[0]      = e0 * inv;
    op[HW]     = e1 * inv;
    op[2 * HW] = e2 * inv;
}

extern "C" void kernel_launch(void* const* d_in, const int* in_sizes, int n_in,
                              void* d_out, int out_size, void* d_ws, size_t ws_size,
                              hipStream_t stream)
{
    (void)in_sizes; (void)n_in; (void)out_size; (void)d_ws; (void)ws_size;

    const float* V = (const float*)d_in[0];

    const int npix = Bb * (int)HW;  // 262144

    // d_out: [visible_class : npix x int32][visible_probs : B*C*H*W x fp32]
    int*   out_cls   = (int*)d_out;
    float* out_probs = (float*)((char*)d_out + (size_t)npix * sizeof(int));

    dim3 block(256);
    dim3 grid((unsigned)(HW / 256), (unsigned)Bb);  // 256 x 4 blocks, exact
    hipLaunchKernelGGL(first_hit_projector_kernel, grid, block, 0, stream,
                       V, out_cls, out_probs);
}